// GCNEncoder_3693671874794
// MI455X (gfx1250) — compile-verified
//
#include <hip/hip_runtime.h>
#include <cmath>

#define NNODES 100000
#define NEDGES 600000
#define FEAT   128
#define NGRAPH 256

typedef __attribute__((ext_vector_type(2))) float v2f;
typedef __attribute__((ext_vector_type(8))) float v8f;

// ---------------- init: zero pooled output + counts, deg = 1 (self loop) ----
__global__ void gcn_init_kernel(float* __restrict__ out, float* __restrict__ cnts,
                                float* __restrict__ deg) {
    int i = blockIdx.x * 256 + threadIdx.x;
    if (i < NGRAPH * FEAT) out[i] = 0.0f;
    if (i < NGRAPH)        cnts[i] = 0.0f;
    if (i < NNODES)        deg[i]  = 1.0f;   // self loop contributes 1 to degree
}

// ---------------- degree accumulation over col ------------------------------
__global__ void gcn_deg_kernel(const int* __restrict__ col, float* __restrict__ deg) {
    int e = blockIdx.x * 256 + threadIdx.x;
    if (e < NEDGES) atomicAdd(&deg[col[e]], 1.0f);
}

// ---------------- deg -> deg^-0.5 in place ----------------------------------
__global__ void gcn_dinv_kernel(float* __restrict__ deg) {
    int n = blockIdx.x * 256 + threadIdx.x;
    if (n < NNODES) {
        float d = deg[n];
        deg[n] = (d > 0.0f) ? rsqrtf(d) : 0.0f;
    }
}

// ---------------- dense Y = (relu?)X @ W + b via fp32 WMMA ------------------
// block = 256 threads = 8 waves; each wave owns a 16-row M tile.
// Outer loop over 8 column tiles of 16; W column-panel staged in LDS (8 KB).
template <bool RELU>
__global__ void gcn_gemm_kernel(const float* __restrict__ X,
                                const float* __restrict__ W,
                                const float* __restrict__ bias,
                                float* __restrict__ Y) {
    __shared__ float Wt[FEAT * 16];   // W[k][nt*16 + c] panel
    __shared__ float Bt[16];

    const int lane  = threadIdx.x & 31;
    const int wave  = threadIdx.x >> 5;
    const int m0    = blockIdx.x * 128 + wave * 16;
    const bool active = (m0 < NNODES);          // wave-uniform (16 | NNODES)

    const int mrow  = m0 + (lane & 15);         // A row for this lane
    const int khalf = (lane >> 4) << 1;         // 0 (lanes 0-15) or 2 (lanes 16-31)
    const int ncol  = lane & 15;                // B/D column for this lane
    const int half  = lane >> 4;                // D row offset selector

    for (int nt = 0; nt < 8; ++nt) {
        __syncthreads();
        // cooperative load of 128x16 panel: 2048 floats / 256 threads = 8 each
        #pragma unroll
        for (int i = 0; i < 8; ++i) {
            int idx = threadIdx.x * 8 + i;      // 0..2047
            int kk  = idx >> 4;                 // 0..127
            int cc  = idx & 15;
            Wt[idx] = W[kk * FEAT + nt * 16 + cc];
        }
        if (threadIdx.x < 16) Bt[threadIdx.x] = bias[nt * 16 + threadIdx.x];
        __syncthreads();

        if (!active) continue;                  // still hits both barriers next iter

        v8f acc = {};
        #pragma unroll 4
        for (int k = 0; k < FEAT; k += 4) {
            float a0 = X[(size_t)mrow * FEAT + k + khalf];
            float a1 = X[(size_t)mrow * FEAT + k + khalf + 1];
            if (RELU) { a0 = fmaxf(a0, 0.0f); a1 = fmaxf(a1, 0.0f); }
            v2f a; a[0] = a0; a[1] = a1;
            v2f b;
            b[0] = Wt[(k + khalf) * 16 + ncol];
            b[1] = Wt[(k + khalf + 1) * 16 + ncol];
            // D = A(16x4,f32) * B(4x16,f32) + C(16x16,f32)
            acc = __builtin_amdgcn_wmma_f32_16x16x4_f32(
                false, a, false, b, (short)0, acc, false, false);
        }

        // C/D layout: VGPR v -> row m0+v (lanes 0-15) / m0+v+8 (lanes 16-31)
        #pragma unroll
        for (int v = 0; v < 8; ++v) {
            int r = m0 + v + half * 8;
            Y[(size_t)r * FEAT + nt * 16 + ncol] = acc[v] + Bt[ncol];
        }
    }
}

// ---------------- AGG[n] = dinv[n]^2 * H[n]  (self-loop message, no atomics) -
__global__ void gcn_agg_init_kernel(const float* __restrict__ H,
                                    const float* __restrict__ dinv,
                                    float* __restrict__ AGG) {
    int i = blockIdx.x * 256 + threadIdx.x;     // over NNODES*32 quads
    if (i >= NNODES * 32) return;
    int n = i >> 5;
    float s = dinv[n]; s = s * s;
    float4 h = *(const float4*)(H + (size_t)i * 4);
    float4 r = make_float4(h.x * s, h.y * s, h.z * s, h.w * s);
    *(float4*)(AGG + (size_t)i * 4) = r;
}

// ---------------- edge scatter: AGG[row] += norm * H[col] -------------------
// one wave per edge; each lane handles 4 contiguous floats
__global__ void gcn_scatter_kernel(const int* __restrict__ row,
                                   const int* __restrict__ col,
                                   const float* __restrict__ dinv,
                                   const float* __restrict__ H,
                                   float* __restrict__ AGG) {
    int t = blockIdx.x * 256 + threadIdx.x;
    int e = t >> 5;
    if (e >= NEDGES) return;
    int lane = t & 31;
    int r = row[e];
    int c = col[e];
    float norm = dinv[r] * dinv[c];
    float4 h = *(const float4*)(H + (size_t)c * FEAT + lane * 4);
    float* dst = AGG + (size_t)r * FEAT + lane * 4;
    atomicAdd(dst + 0, norm * h.x);
    atomicAdd(dst + 1, norm * h.y);
    atomicAdd(dst + 2, norm * h.z);
    atomicAdd(dst + 3, norm * h.w);
}

// ---------------- mean pool: sums into out, counts into cnts ----------------
__global__ void gcn_pool_kernel(const float* __restrict__ AGG,
                                const int* __restrict__ batch,
                                float* __restrict__ out,
                                float* __restrict__ cnts) {
    int t = blockIdx.x * 256 + threadIdx.x;
    int n = t >> 5;
    if (n >= NNODES) return;
    int lane = t & 31;
    int g = batch[n];
    float4 h = *(const float4*)(AGG + (size_t)n * FEAT + lane * 4);
    float* dst = out + (size_t)g * FEAT + lane * 4;
    atomicAdd(dst + 0, h.x);
    atomicAdd(dst + 1, h.y);
    atomicAdd(dst + 2, h.z);
    atomicAdd(dst + 3, h.w);
    if (lane == 0) atomicAdd(&cnts[g], 1.0f);
}

__global__ void gcn_finalize_kernel(float* __restrict__ out,
                                    const float* __restrict__ cnts) {
    int i = blockIdx.x * 256 + threadIdx.x;
    if (i < NGRAPH * FEAT) {
        float c = cnts[i >> 7];                 // FEAT == 128
        out[i] = out[i] / fmaxf(c, 1.0f);
    }
}

extern "C" void kernel_launch(void* const* d_in, const int* in_sizes, int n_in,
                              void* d_out, int out_size, void* d_ws, size_t ws_size,
                              hipStream_t stream) {
    const float* x          = (const float*)d_in[0];
    const int*   edge_index = (const int*)  d_in[1];
    const int*   batch      = (const int*)  d_in[2];
    // d_in[3] = num_graphs scalar (compile-time NGRAPH)
    const float* W1 = (const float*)d_in[4];
    const float* b1 = (const float*)d_in[5];
    const float* W2 = (const float*)d_in[6];
    const float* b2 = (const float*)d_in[7];
    float* out = (float*)d_out;

    // workspace layout
    float* H    = (float*)d_ws;                       // N*128  (H1, then H2)
    float* AGG  = H    + (size_t)NNODES * FEAT;       // N*128  (AGG1, then AGG2)
    float* dinv = AGG  + (size_t)NNODES * FEAT;       // N      (deg, then deg^-1/2)
    float* cnts = dinv + NNODES;                      // NGRAPH

    const int* row = edge_index;                      // edge_index[0]
    const int* col = edge_index + NEDGES;             // edge_index[1]

    dim3 blk(256);
    const int grid_nodes  = (NNODES + 255) / 256;
    const int grid_edges  = (NEDGES + 255) / 256;
    const int grid_gemm   = (NNODES + 127) / 128;
    const int grid_nfeat4 = (NNODES * 32 + 255) / 256;   // node x 32 lanes
    const int grid_escat  = (NEDGES * 32 + 255) / 256;   // edge x 32 lanes
    const int grid_out    = (NGRAPH * FEAT + 255) / 256;

    // normalization
    gcn_init_kernel<<<grid_nodes, blk, 0, stream>>>(out, cnts, dinv);
    gcn_deg_kernel<<<grid_edges, blk, 0, stream>>>(col, dinv);
    gcn_dinv_kernel<<<grid_nodes, blk, 0, stream>>>(dinv);

    // layer 1: H1 = x @ W1 + b1 ; AGG1 = D^-1/2 A D^-1/2 H1
    gcn_gemm_kernel<false><<<grid_gemm, blk, 0, stream>>>(x, W1, b1, H);
    gcn_agg_init_kernel<<<grid_nfeat4, blk, 0, stream>>>(H, dinv, AGG);
    gcn_scatter_kernel<<<grid_escat, blk, 0, stream>>>(row, col, dinv, H, AGG);

    // layer 2: H2 = relu(AGG1) @ W2 + b2 ; AGG2 = D^-1/2 A D^-1/2 H2
    gcn_gemm_kernel<true><<<grid_gemm, blk, 0, stream>>>(AGG, W2, b2, H);
    gcn_agg_init_kernel<<<grid_nfeat4, blk, 0, stream>>>(H, dinv, AGG);
    gcn_scatter_kernel<<<grid_escat, blk, 0, stream>>>(row, col, dinv, H, AGG);

    // global mean pool
    gcn_pool_kernel<<<grid_nfeat4, blk, 0, stream>>>(AGG, batch, out, cnts);
    gcn_finalize_kernel<<<grid_out, blk, 0, stream>>>(out, cnts);
}